// Actor_52269751992940
// MI455X (gfx1250) — compile-verified
//
#include <hip/hip_runtime.h>
#include <math.h>

#define N_ATOMS 8192
#define N_MOLS 128
#define APMOL 64
#define NB 128
#define NRBF 50
#define CUT 5.0f
#define PI_F 3.14159265358979f
#define DELTA (CUT / 49.0f)
#define RBF_COEFF (-0.5f / (DELTA * DELTA))
#define LIMITF 1.0f
#define EPSF 1e-8f

typedef __attribute__((ext_vector_type(2))) float v2f;
typedef __attribute__((ext_vector_type(8))) float v8f;

__device__ __forceinline__ float sspf(float x) {
    float sp = (x > 20.f) ? x : log1pf(expf(x));
    return sp - 0.6931471805599453f;
}
__device__ __forceinline__ float sigf(float x) { return 1.f / (1.f + expf(-x)); }

__device__ __forceinline__ v8f wmma4(v2f a, v2f b, v8f c) {
    // D = A(16x4, f32) * B(4x16, f32) + C(16x16, f32)
    return __builtin_amdgcn_wmma_f32_16x16x4_f32(false, a, false, b, (short)0, c,
                                                 false, false);
}

// ---------------------------------------------------------------------------
// x0[n,:] = embedding[Z[n],:]
// ---------------------------------------------------------------------------
__global__ __launch_bounds__(128) void embed_kernel(const int* __restrict__ Z,
                                                    const float* __restrict__ emb,
                                                    float* __restrict__ x) {
    int n = blockIdx.x;
    int c = threadIdx.x;
    x[n * NB + c] = emb[Z[n] * NB + c];
}

// ---------------------------------------------------------------------------
// Generic C[M,128] = act(A[M,128] @ op(B) + bias) (+ addsrc)
// transB: 0 -> B[k,n] ; 1 -> B[n,k]
// act: 0 none, 1 shifted-softplus, 2 multiply by sigmoid(sigsrc)
// grid.x = M/16, block = 256 (8 waves; wave w owns cols [16w,16w+16))
// ---------------------------------------------------------------------------
__global__ __launch_bounds__(256) void gemm128_kernel(
    const float* __restrict__ A, const float* __restrict__ B,
    const float* __restrict__ bias, const float* __restrict__ addsrc,
    const float* __restrict__ sigsrc, float* __restrict__ C, int transB, int act) {
    __shared__ float sA[16][NB];
    const int tid = threadIdx.x;
    const int wave = tid >> 5, lane = tid & 31, hi = lane >> 4, ln = lane & 15;
    const int m0 = blockIdx.x * 16;

    for (int idx = tid; idx < 16 * NB; idx += 256) {
        int r = idx >> 7, c = idx & 127;
        sA[r][c] = A[(m0 + r) * NB + c];
    }
    __syncthreads();

    const int col = wave * 16 + ln;
    v8f acc = {0.f, 0.f, 0.f, 0.f, 0.f, 0.f, 0.f, 0.f};
    if (!transB) {
        for (int k = 0; k < NB; k += 4) {
            int k0 = k + 2 * hi;
            v2f a, b;
            a.x = sA[ln][k0];
            a.y = sA[ln][k0 + 1];
            b.x = B[k0 * NB + col];
            b.y = B[(k0 + 1) * NB + col];
            acc = wmma4(a, b, acc);
        }
    } else {
        for (int k = 0; k < NB; k += 4) {
            int k0 = k + 2 * hi;
            v2f a, b;
            a.x = sA[ln][k0];
            a.y = sA[ln][k0 + 1];
            b.x = B[col * NB + k0];
            b.y = B[col * NB + k0 + 1];
            acc = wmma4(a, b, acc);
        }
    }
    float bc = bias ? bias[col] : 0.f;
#pragma unroll
    for (int v = 0; v < 8; ++v) {
        int m = m0 + v + 8 * hi;
        float z = acc[v] + bc;
        if (act == 1)
            z = sspf(z);
        else if (act == 2)
            z *= sigf(sigsrc[m * NB + col]);
        if (addsrc) z += addsrc[m * NB + col];
        C[m * NB + col] = z;
    }
}

// ---------------------------------------------------------------------------
// Fused pair filter + message + aggregate (forward), 16 pairs per block.
// W = ssp(rbf@fW1+b1)@fW2+b2, scaled by fcut; agg[i] += h[j]*W (atomics)
// ---------------------------------------------------------------------------
__global__ __launch_bounds__(256) void pair_fwd_kernel(
    const float* __restrict__ pos, const int* __restrict__ idx_i,
    const int* __restrict__ idx_j, const float* __restrict__ fW1,
    const float* __restrict__ fb1, const float* __restrict__ fW2,
    const float* __restrict__ fb2, const float* __restrict__ h,
    float* __restrict__ agg, int P) {
    __shared__ float sD[16], sFc[16];
    __shared__ int sI[16], sJ[16], sV[16];
    __shared__ float sRbf[16][52];
    __shared__ float sS1[16][NB];

    const int tid = threadIdx.x;
    const int wave = tid >> 5, lane = tid & 31, hi = lane >> 4, ln = lane & 15;
    const int p0 = blockIdx.x * 16;

    if (tid < 16) {
        int pp = p0 + tid;
        int valid = pp < P ? 1 : 0;
        int p = valid ? pp : (P - 1);
        int i = idx_i[p], j = idx_j[p];
        sI[tid] = i;
        sJ[tid] = j;
        sV[tid] = valid;
        float rx = pos[j * 3 + 0] - pos[i * 3 + 0];
        float ry = pos[j * 3 + 1] - pos[i * 3 + 1];
        float rz = pos[j * 3 + 2] - pos[i * 3 + 2];
        float d = sqrtf(rx * rx + ry * ry + rz * rz);
        sD[tid] = d;
        float fc = (d < CUT) ? 0.5f * (cosf(PI_F * d / CUT) + 1.f) : 0.f;
        sFc[tid] = valid ? fc : 0.f;
    }
    __syncthreads();
    for (int idx = tid; idx < 16 * 52; idx += 256) {
        int m = idx / 52, k = idx % 52;
        float v = 0.f;
        if (k < NRBF) {
            float t = sD[m] - k * DELTA;
            v = expf(RBF_COEFF * t * t);
        }
        sRbf[m][k] = v;
    }
    __syncthreads();

    const int col = wave * 16 + ln;
    // GEMM1: a1 = rbf @ fW1  (K = 50 padded to 52)
    v8f acc1 = {0.f, 0.f, 0.f, 0.f, 0.f, 0.f, 0.f, 0.f};
    for (int k = 0; k < 52; k += 4) {
        int k0 = k + 2 * hi;
        v2f a, b;
        a.x = sRbf[ln][k0];
        a.y = sRbf[ln][k0 + 1];
        b.x = (k0 < NRBF) ? fW1[k0 * NB + col] : 0.f;
        b.y = (k0 + 1 < NRBF) ? fW1[(k0 + 1) * NB + col] : 0.f;
        acc1 = wmma4(a, b, acc1);
    }
    float b1 = fb1[col];
#pragma unroll
    for (int v = 0; v < 8; ++v) {
        int m = v + 8 * hi;
        sS1[m][col] = sspf(acc1[v] + b1);
    }
    __syncthreads();

    // GEMM2: G = s1 @ fW2  (K = 128)
    v8f acc2 = {0.f, 0.f, 0.f, 0.f, 0.f, 0.f, 0.f, 0.f};
    for (int k = 0; k < NB; k += 4) {
        int k0 = k + 2 * hi;
        v2f a, b;
        a.x = sS1[ln][k0];
        a.y = sS1[ln][k0 + 1];
        b.x = fW2[k0 * NB + col];
        b.y = fW2[(k0 + 1) * NB + col];
        acc2 = wmma4(a, b, acc2);
    }
    float b2 = fb2[col];
#pragma unroll
    for (int v = 0; v < 8; ++v) {
        int m = v + 8 * hi;
        float W = (acc2[v] + b2) * sFc[m];  // sFc == 0 for invalid rows
        int j = sJ[m], i = sI[m];
        float msg = h[j * NB + col] * W;
        atomicAdd(&agg[i * NB + col], msg);
    }
}

// ---------------------------------------------------------------------------
// Fused pair backward: recompute filter, scatter grad_h and grad_pos.
// ---------------------------------------------------------------------------
__global__ __launch_bounds__(256) void pair_bwd_kernel(
    const float* __restrict__ pos, const int* __restrict__ idx_i,
    const int* __restrict__ idx_j, const float* __restrict__ fW1,
    const float* __restrict__ fb1, const float* __restrict__ fW2,
    const float* __restrict__ fb2, const float* __restrict__ h,
    const float* __restrict__ gagg, float* __restrict__ gh,
    float* __restrict__ gpos, int P) {
    __shared__ float sD[16], sFc[16], sDfc[16], sRx[16], sRy[16], sRz[16];
    __shared__ float sGd[16], sGfc[16];
    __shared__ int sI[16], sJ[16], sV[16];
    __shared__ float sRbf[16][52];
    __shared__ float sDrbf[16][52];
    __shared__ float sS1[16][NB];
    __shared__ float sSig1[16][NB];
    __shared__ float sGG[16][NB];

    const int tid = threadIdx.x;
    const int wave = tid >> 5, lane = tid & 31, hi = lane >> 4, ln = lane & 15;
    const int p0 = blockIdx.x * 16;

    if (tid < 16) {
        int pp = p0 + tid;
        int valid = pp < P ? 1 : 0;
        int p = valid ? pp : (P - 1);
        int i = idx_i[p], j = idx_j[p];
        sI[tid] = i;
        sJ[tid] = j;
        sV[tid] = valid;
        float rx = pos[j * 3 + 0] - pos[i * 3 + 0];
        float ry = pos[j * 3 + 1] - pos[i * 3 + 1];
        float rz = pos[j * 3 + 2] - pos[i * 3 + 2];
        float d = sqrtf(rx * rx + ry * ry + rz * rz);
        sRx[tid] = rx;
        sRy[tid] = ry;
        sRz[tid] = rz;
        sD[tid] = d;
        bool in = d < CUT;
        sFc[tid] = (valid && in) ? 0.5f * (cosf(PI_F * d / CUT) + 1.f) : 0.f;
        sDfc[tid] = (valid && in) ? (-0.5f * PI_F / CUT) * sinf(PI_F * d / CUT) : 0.f;
        sGd[tid] = 0.f;
        sGfc[tid] = 0.f;
    }
    __syncthreads();
    for (int idx = tid; idx < 16 * 52; idx += 256) {
        int m = idx / 52, k = idx % 52;
        float rv = 0.f, dv = 0.f;
        if (k < NRBF) {
            float t = sD[m] - k * DELTA;
            rv = expf(RBF_COEFF * t * t);
            dv = rv * 2.f * RBF_COEFF * t;  // d rbf / d d
        }
        sRbf[m][k] = rv;
        sDrbf[m][k] = dv;
    }
    __syncthreads();

    const int col = wave * 16 + ln;
    // GEMM1: a1 = rbf @ fW1 + b1 ; keep ssp(a1) and sigmoid(a1)
    v8f acc1 = {0.f, 0.f, 0.f, 0.f, 0.f, 0.f, 0.f, 0.f};
    for (int k = 0; k < 52; k += 4) {
        int k0 = k + 2 * hi;
        v2f a, b;
        a.x = sRbf[ln][k0];
        a.y = sRbf[ln][k0 + 1];
        b.x = (k0 < NRBF) ? fW1[k0 * NB + col] : 0.f;
        b.y = (k0 + 1 < NRBF) ? fW1[(k0 + 1) * NB + col] : 0.f;
        acc1 = wmma4(a, b, acc1);
    }
    float b1 = fb1[col];
#pragma unroll
    for (int v = 0; v < 8; ++v) {
        int m = v + 8 * hi;
        float a1 = acc1[v] + b1;
        sS1[m][col] = sspf(a1);
        sSig1[m][col] = sigf(a1);
    }
    __syncthreads();

    // GEMM2: G = s1 @ fW2 + b2
    v8f accG = {0.f, 0.f, 0.f, 0.f, 0.f, 0.f, 0.f, 0.f};
    for (int k = 0; k < NB; k += 4) {
        int k0 = k + 2 * hi;
        v2f a, b;
        a.x = sS1[ln][k0];
        a.y = sS1[ln][k0 + 1];
        b.x = fW2[k0 * NB + col];
        b.y = fW2[(k0 + 1) * NB + col];
        accG = wmma4(a, b, accG);
    }
    float b2 = fb2[col];
#pragma unroll
    for (int v = 0; v < 8; ++v) {
        int m = v + 8 * hi;
        float G = accG[v] + b2;
        int i = sI[m], j = sJ[m];
        float ga = gagg[i * NB + col];
        float fc = sFc[m];
        float W = G * fc;
        atomicAdd(&gh[j * NB + col], ga * W);  // grad h[j] += grad_msg * W
        float hj = h[j * NB + col];
        float gW = ga * hj;       // dL/dW
        atomicAdd(&sGfc[m], gW * G);  // grad fcut partial
        sGG[m][col] = gW * fc;    // grad G
    }
    __syncthreads();

    // GEMM3: grad_s1 = grad_G @ fW2^T ; grad_a1 = grad_s1 * sigmoid(a1)
    v8f accS = {0.f, 0.f, 0.f, 0.f, 0.f, 0.f, 0.f, 0.f};
    for (int k = 0; k < NB; k += 4) {
        int k0 = k + 2 * hi;
        v2f a, b;
        a.x = sGG[ln][k0];
        a.y = sGG[ln][k0 + 1];
        b.x = fW2[col * NB + k0];
        b.y = fW2[col * NB + k0 + 1];
        accS = wmma4(a, b, accS);
    }
    float gA1v[8];
#pragma unroll
    for (int v = 0; v < 8; ++v) {
        int m = v + 8 * hi;
        gA1v[v] = accS[v] * sSig1[m][col];
    }

    // GEMM4: tt = drbf_dd @ fW1 ; grad_d[m] += sum_col grad_a1 * tt
    v8f accT = {0.f, 0.f, 0.f, 0.f, 0.f, 0.f, 0.f, 0.f};
    for (int k = 0; k < 52; k += 4) {
        int k0 = k + 2 * hi;
        v2f a, b;
        a.x = sDrbf[ln][k0];
        a.y = sDrbf[ln][k0 + 1];
        b.x = (k0 < NRBF) ? fW1[k0 * NB + col] : 0.f;
        b.y = (k0 + 1 < NRBF) ? fW1[(k0 + 1) * NB + col] : 0.f;
        accT = wmma4(a, b, accT);
    }
#pragma unroll
    for (int v = 0; v < 8; ++v) {
        int m = v + 8 * hi;
        atomicAdd(&sGd[m], gA1v[v] * accT[v]);
    }
    __syncthreads();

    if (tid < 16 && sV[tid]) {
        int m = tid;
        float gd = sGd[m] + sGfc[m] * sDfc[m];
        float d = sD[m];
        float s = (d > 1e-10f) ? gd / d : 0.f;
        float gx = s * sRx[m], gy = s * sRy[m], gz = s * sRz[m];
        int j = sJ[m], i = sI[m];
        atomicAdd(&gpos[j * 3 + 0], gx);
        atomicAdd(&gpos[j * 3 + 1], gy);
        atomicAdd(&gpos[j * 3 + 2], gz);
        atomicAdd(&gpos[i * 3 + 0], -gx);
        atomicAdd(&gpos[i * 3 + 1], -gy);
        atomicAdd(&gpos[i * 3 + 2], -gz);
    }
}

// ---------------------------------------------------------------------------
// Output head: e = ssp(x@aW1+ab1)@aW2+ab2; energy[mol] += e;
// seed grad_x = (aW2*sigmoid(t1)) @ aW1^T. One atom per 64-thread block.
// ---------------------------------------------------------------------------
__global__ __launch_bounds__(64) void head_kernel(
    const float* __restrict__ x, const float* __restrict__ aW1,
    const float* __restrict__ ab1, const float* __restrict__ aW2,
    const float* __restrict__ ab2, const int* __restrict__ mol,
    float* __restrict__ energy, float* __restrict__ gx) {
    int atom = blockIdx.x;
    int c = threadIdx.x;  // 0..63
    __shared__ float sx[NB];
    __shared__ float sg[64];
    __shared__ float sE;
    sx[c] = x[atom * NB + c];
    sx[c + 64] = x[atom * NB + c + 64];
    if (c == 0) sE = 0.f;
    __syncthreads();
    float t1 = ab1[c];
#pragma unroll 8
    for (int k = 0; k < NB; ++k) t1 += sx[k] * aW1[k * 64 + c];
    float s = sspf(t1);
    float w2 = aW2[c];
    atomicAdd(&sE, s * w2);
    sg[c] = w2 * sigf(t1);
    __syncthreads();
    if (c == 0) atomicAdd(&energy[mol[atom]], sE + ab2[0]);
    for (int k = c; k < NB; k += 64) {
        float g = 0.f;
#pragma unroll 8
        for (int cc = 0; cc < 64; ++cc) g += sg[cc] * aW1[k * 64 + cc];
        gx[atom * NB + k] = g;
    }
}

// ---------------------------------------------------------------------------
// forces = -grad_pos; per-molecule max-norm clip; write action.
// ---------------------------------------------------------------------------
__global__ __launch_bounds__(64) void finalize_kernel(const float* __restrict__ gpos,
                                                      float* __restrict__ action) {
    int m = blockIdx.x, a = threadIdx.x;
    int atom = m * APMOL + a;
    float fx = -gpos[atom * 3 + 0];
    float fy = -gpos[atom * 3 + 1];
    float fz = -gpos[atom * 3 + 2];
    float nr = sqrtf(fx * fx + fy * fy + fz * fz);
    __shared__ float red[64];
    red[a] = nr;
    __syncthreads();
    for (int s = 32; s > 0; s >>= 1) {
        if (a < s) red[a] = fmaxf(red[a], red[a + s]);
        __syncthreads();
    }
    float mx = fmaxf(red[0], EPSF);
    float coef = fminf(LIMITF / mx, 1.f);
    action[atom * 3 + 0] = fx * coef;
    action[atom * 3 + 1] = fy * coef;
    action[atom * 3 + 2] = fz * coef;
}

// ---------------------------------------------------------------------------
extern "C" void kernel_launch(void* const* d_in, const int* in_sizes, int n_in,
                              void* d_out, int out_size, void* d_ws, size_t ws_size,
                              hipStream_t stream) {
    const float* pos   = (const float*)d_in[0];
    const int*   Z     = (const int*)d_in[1];
    const int*   idx_i = (const int*)d_in[2];
    const int*   idx_j = (const int*)d_in[3];
    const int*   mol   = (const int*)d_in[4];
    const float* emb   = (const float*)d_in[5];
    const float* fW1   = (const float*)d_in[6];
    const float* fb1   = (const float*)d_in[7];
    const float* fW2   = (const float*)d_in[8];
    const float* fb2   = (const float*)d_in[9];
    const float* in2fW = (const float*)d_in[10];
    const float* in2fb = (const float*)d_in[11];
    const float* o1W   = (const float*)d_in[12];
    const float* o1b   = (const float*)d_in[13];
    const float* o2W   = (const float*)d_in[14];
    const float* o2b   = (const float*)d_in[15];
    const float* aW1   = (const float*)d_in[16];
    const float* ab1   = (const float*)d_in[17];
    const float* aW2   = (const float*)d_in[18];
    const float* ab2   = (const float*)d_in[19];

    const int P = in_sizes[2];
    const int ptiles = (P + 15) / 16;

    float* action = (float*)d_out;                 // [N_ATOMS*3]
    float* energy = (float*)d_out + N_ATOMS * 3;   // [N_MOLS]

    const size_t NF = (size_t)N_ATOMS * NB;
    float* w = (float*)d_ws;
    float* x[4];
    for (int t = 0; t < 4; ++t) x[t] = w + t * NF;
    float* aggb[3];
    for (int t = 0; t < 3; ++t) aggb[t] = w + (4 + t) * NF;
    float* hbuf  = w + 7 * NF;
    float* sbuf  = w + 8 * NF;   // forward s ; reused as grad_u in backward
    float* ubuf  = w + 9 * NF;
    float* gagg  = w + 10 * NF;
    float* gxa   = w + 11 * NF;
    float* gxb   = w + 12 * NF;
    float* ghb   = w + 13 * NF;
    float* gpos  = w + 14 * NF;  // N_ATOMS*3 floats

    const dim3 blk256(256), blk128(128), blk64(64);
    const dim3 gGemm(N_ATOMS / 16), gPair(ptiles), gAtom(N_ATOMS), gMol(N_MOLS);

    // ---------------- forward ----------------
    embed_kernel<<<gAtom, blk128, 0, stream>>>(Z, emb, x[0]);
    for (int t = 0; t < 3; ++t) {
        const float* fW1t = fW1 + (size_t)t * NRBF * NB;
        const float* fb1t = fb1 + (size_t)t * NB;
        const float* fW2t = fW2 + (size_t)t * NB * NB;
        const float* fb2t = fb2 + (size_t)t * NB;
        // h = x @ in2fW + b
        gemm128_kernel<<<gGemm, blk256, 0, stream>>>(
            x[t], in2fW + (size_t)t * NB * NB, in2fb + (size_t)t * NB, nullptr,
            nullptr, hbuf, 0, 0);
        hipMemsetAsync(aggb[t], 0, NF * sizeof(float), stream);
        pair_fwd_kernel<<<gPair, blk256, 0, stream>>>(pos, idx_i, idx_j, fW1t, fb1t,
                                                      fW2t, fb2t, hbuf, aggb[t], P);
        // s = ssp(agg @ o1W + o1b)
        gemm128_kernel<<<gGemm, blk256, 0, stream>>>(
            aggb[t], o1W + (size_t)t * NB * NB, o1b + (size_t)t * NB, nullptr,
            nullptr, sbuf, 0, 1);
        // x_{t+1} = x_t + s @ o2W + o2b
        gemm128_kernel<<<gGemm, blk256, 0, stream>>>(
            sbuf, o2W + (size_t)t * NB * NB, o2b + (size_t)t * NB, x[t], nullptr,
            x[t + 1], 0, 0);
    }

    // ---------------- head (energy + grad seed) ----------------
    hipMemsetAsync(energy, 0, N_MOLS * sizeof(float), stream);
    head_kernel<<<gAtom, blk64, 0, stream>>>(x[3], aW1, ab1, aW2, ab2, mol, energy,
                                             gxa);

    // ---------------- backward ----------------
    hipMemsetAsync(gpos, 0, (size_t)N_ATOMS * 3 * sizeof(float), stream);
    for (int t = 2; t >= 0; --t) {
        const float* fW1t = fW1 + (size_t)t * NRBF * NB;
        const float* fb1t = fb1 + (size_t)t * NB;
        const float* fW2t = fW2 + (size_t)t * NB * NB;
        const float* fb2t = fb2 + (size_t)t * NB;
        // u = agg @ o1W + o1b
        gemm128_kernel<<<gGemm, blk256, 0, stream>>>(
            aggb[t], o1W + (size_t)t * NB * NB, o1b + (size_t)t * NB, nullptr,
            nullptr, ubuf, 0, 0);
        // grad_u = (grad_x @ o2W^T) * sigmoid(u)
        gemm128_kernel<<<gGemm, blk256, 0, stream>>>(
            gxa, o2W + (size_t)t * NB * NB, nullptr, nullptr, ubuf, sbuf, 1, 2);
        // grad_agg = grad_u @ o1W^T
        gemm128_kernel<<<gGemm, blk256, 0, stream>>>(
            sbuf, o1W + (size_t)t * NB * NB, nullptr, nullptr, nullptr, gagg, 1, 0);
        // h = x_t @ in2fW + b  (recompute)
        gemm128_kernel<<<gGemm, blk256, 0, stream>>>(
            x[t], in2fW + (size_t)t * NB * NB, in2fb + (size_t)t * NB, nullptr,
            nullptr, hbuf, 0, 0);
        hipMemsetAsync(ghb, 0, NF * sizeof(float), stream);
        pair_bwd_kernel<<<gPair, blk256, 0, stream>>>(pos, idx_i, idx_j, fW1t, fb1t,
                                                      fW2t, fb2t, hbuf, gagg, ghb,
                                                      gpos, P);
        // grad_x_prev = grad_h @ in2fW^T + grad_x (residual)
        gemm128_kernel<<<gGemm, blk256, 0, stream>>>(
            ghb, in2fW + (size_t)t * NB * NB, nullptr, gxa, nullptr, gxb, 1, 0);
        float* tmp = gxa;
        gxa = gxb;
        gxb = tmp;
    }

    // ---------------- clip + output ----------------
    finalize_kernel<<<gMol, blk64, 0, stream>>>(gpos, action);
}